// QGCN_44289702756373
// MI455X (gfx1250) — compile-verified
//
#include <hip/hip_runtime.h>

typedef __attribute__((ext_vector_type(2))) float v2f;
typedef __attribute__((ext_vector_type(8))) float v8f;

#define NN 100000
#define NE 1600000
#define F 128
#define NC 40
#define WPAD 132   // LDS row pitch in words: 132 % 64 == 4 -> conflict-free fragment reads

// ---------------- degrees & norms ----------------
__global__ void __launch_bounds__(256) k_degree(const int* __restrict__ src,
                                                const int* __restrict__ dst,
                                                float* __restrict__ degs,
                                                float* __restrict__ degd) {
  int e = blockIdx.x * 256 + threadIdx.x;
  if (e >= NE) return;
  atomicAdd(&degs[src[e]], 1.0f);
  atomicAdd(&degd[dst[e]], 1.0f);
}

__global__ void __launch_bounds__(256) k_norm(float* __restrict__ buf) {
  int i = blockIdx.x * 256 + threadIdx.x;
  if (i >= 2 * NN) return;
  buf[i] = rsqrtf(fmaxf(buf[i], 1.0f));
}

// ---------------- row scale: hs = h * out_norm ----------------
__global__ void __launch_bounds__(256) k_scale(const float* __restrict__ h,
                                               const float* __restrict__ nrm,
                                               float* __restrict__ hs) {
  int i = blockIdx.x * 256 + threadIdx.x;     // one float4 per thread
  if (i >= NN * (F / 4)) return;
  int off = i * 4;
  int row = off >> 7;
  float4 v = *(const float4*)(h + off);
  float s = nrm[row];
  v.x *= s; v.y *= s; v.z *= s; v.w *= s;
  *(float4*)(hs + off) = v;
}

// ---------------- SpMM scatter-add, 128 wide: one wave per edge ----------------
__global__ void __launch_bounds__(256) k_spmm128(const float* __restrict__ hs,
                                                 const int* __restrict__ src,
                                                 const int* __restrict__ dst,
                                                 float* __restrict__ agg) {
  long gt = (long)blockIdx.x * 256 + threadIdx.x;
  int e = (int)(gt >> 5);
  int lane = threadIdx.x & 31;
  if (e >= NE) return;
  int s = src[e], d = dst[e];
  float4 v = *(const float4*)(hs + (long)s * F + lane * 4);
  float* o = agg + (long)d * F + lane * 4;
  atomicAdd(o + 0, v.x); atomicAdd(o + 1, v.y);
  atomicAdd(o + 2, v.z); atomicAdd(o + 3, v.w);
}

// ------------- fused z = (agg @ W^T)*in_norm + h @ L^T  (128x128 weights) -------------
// weights staged once per workgroup into padded LDS; one wave per 16-row stripe;
// 8 N-tiles; fp32 WMMA 16x16x4, K=128 -> 32 steps
__global__ void __launch_bounds__(256) k_gemm_z(const float* __restrict__ agg,
                                                const float* __restrict__ h,
                                                const float* __restrict__ W,
                                                const float* __restrict__ Lw,
                                                const float* __restrict__ innorm,
                                                float* __restrict__ z) {
  __shared__ float sW[F * WPAD];
  __shared__ float sL[F * WPAD];
  // cooperative stage: 128x128 floats each, float4 global loads -> ds_store_b128
  for (int i = threadIdx.x; i < F * (F / 4); i += 256) {
    int row = i >> 5;
    int c4 = (i & 31) << 2;
    *(float4*)&sW[row * WPAD + c4] = *(const float4*)(W + row * F + c4);
    *(float4*)&sL[row * WPAD + c4] = *(const float4*)(Lw + row * F + c4);
  }
  __syncthreads();

  const int lane = threadIdx.x & 31;
  const int wave = threadIdx.x >> 5;
  const int rb = blockIdx.x * 8 + wave;
  if (rb * 16 >= NN) return;                  // wave-uniform; after barrier
  const int m0 = rb * 16;
  const int nlan = lane & 15;
  const int koff = (lane >> 4) * 2;           // ISA A/B layout: lanes 16-31 hold k+2,k+3
  const int mA = m0 + nlan;
  const float rn = innorm[mA];                // in_norm folded into A rows of agg
  const float* arow = agg + (long)mA * F + koff;
  const float* hrow = h + (long)mA * F + koff;
  v8f acc[8] = {};
  for (int ks = 0; ks < 32; ++ks) {
    int k0 = ks * 4;
    float2 a1f = *(const float2*)(arow + k0);
    float2 a2f = *(const float2*)(hrow + k0);
    v2f a1; a1.x = a1f.x * rn; a1.y = a1f.y * rn;
    v2f a2; a2.x = a2f.x;      a2.y = a2f.y;
#pragma unroll
    for (int t = 0; t < 8; ++t) {
      int wb = ((t << 4) + nlan) * WPAD + koff + k0;  // W^T(k,n) = W[n][k]
      float2 bwf = *(const float2*)&sW[wb];
      float2 blf = *(const float2*)&sL[wb];
      v2f bw; bw.x = bwf.x; bw.y = bwf.y;
      v2f bl; bl.x = blf.x; bl.y = blf.y;
      acc[t] = __builtin_amdgcn_wmma_f32_16x16x4_f32(false, a1, false, bw,
                                                     (short)0, acc[t], false, false);
      acc[t] = __builtin_amdgcn_wmma_f32_16x16x4_f32(false, a2, false, bl,
                                                     (short)0, acc[t], false, false);
    }
  }
  const int hi8 = (lane >> 4) << 3;           // C layout: lanes>=16 hold M=r+8
#pragma unroll
  for (int t = 0; t < 8; ++t) {
    int col = (t << 4) + nlan;
#pragma unroll
    for (int r = 0; r < 8; ++r)
      z[(long)(m0 + r + hi8) * F + col] = acc[t][r];
  }
}

// ---------------- per-column sum / sumsq for BatchNorm ----------------
__global__ void __launch_bounds__(128) k_colstats(const float* __restrict__ z,
                                                  float* __restrict__ stats) {
  int c = threadIdx.x;                        // 0..127
  int r0 = blockIdx.x * 250;                  // 400 blocks * 250 rows = 100000
  float s = 0.f, sq = 0.f;
  for (int r = r0; r < r0 + 250; ++r) {
    float v = z[(long)r * F + c];
    s += v; sq += v * v;
  }
  atomicAdd(&stats[c], s);
  atomicAdd(&stats[F + c], sq);
}

__global__ void __launch_bounds__(256) k_bnrelu(const float* __restrict__ z,
                                                const float* __restrict__ stats,
                                                const float* __restrict__ g,
                                                const float* __restrict__ beta,
                                                float* __restrict__ out) {
  long i = (long)blockIdx.x * 256 + threadIdx.x;
  if (i >= (long)NN * F) return;
  int c = (int)(i & (F - 1));
  float mu = stats[c] * (1.0f / NN);
  float var = stats[F + c] * (1.0f / NN) - mu * mu;
  float sc = rsqrtf(var + 1e-5f) * g[c];
  float v = (z[i] - mu) * sc + beta[c];
  out[i] = fmaxf(v, 0.0f);
}

// -------- helper: stage a 40x128 weight into a 48-row zero-padded LDS image --------
__device__ __forceinline__ void stage_w40(const float* __restrict__ Wg, float* sW) {
  for (int i = threadIdx.x; i < NC * (F / 4); i += 256) {   // 1280 float4s
    int row = i >> 5;
    int c4 = (i & 31) << 2;
    *(float4*)&sW[row * WPAD + c4] = *(const float4*)(Wg + row * F + c4);
  }
  for (int i = threadIdx.x; i < 8 * WPAD; i += 256)         // zero rows 40..47
    sW[NC * WPAD + i] = 0.0f;
}

// ---------------- p = hs @ w2^T  (128 -> 40, padded to 3 tiles) ----------------
__global__ void __launch_bounds__(256) k_gemm_p(const float* __restrict__ hs,
                                                const float* __restrict__ W2,
                                                float* __restrict__ p) {
  __shared__ float sW[48 * WPAD];
  stage_w40(W2, sW);
  __syncthreads();

  const int lane = threadIdx.x & 31;
  const int wave = threadIdx.x >> 5;
  const int rb = blockIdx.x * 8 + wave;
  if (rb * 16 >= NN) return;
  const int m0 = rb * 16;
  const int nlan = lane & 15;
  const int koff = (lane >> 4) * 2;
  const float* arow = hs + (long)(m0 + nlan) * F + koff;
  v8f acc[3] = {};
  for (int ks = 0; ks < 32; ++ks) {
    int k0 = ks * 4;
    float2 af = *(const float2*)(arow + k0);
    v2f a; a.x = af.x; a.y = af.y;
#pragma unroll
    for (int t = 0; t < 3; ++t) {
      float2 bf = *(const float2*)&sW[(t * 16 + nlan) * WPAD + koff + k0];
      v2f b; b.x = bf.x; b.y = bf.y;
      acc[t] = __builtin_amdgcn_wmma_f32_16x16x4_f32(false, a, false, b,
                                                     (short)0, acc[t], false, false);
    }
  }
  const int hi8 = (lane >> 4) << 3;
#pragma unroll
  for (int t = 0; t < 3; ++t) {
    int col = t * 16 + nlan;
    if (col < NC) {
#pragma unroll
      for (int r = 0; r < 8; ++r)
        p[(long)(m0 + r + hi8) * NC + col] = acc[t][r];
    }
  }
}

// ---------------- SpMM scatter-add, 40 wide ----------------
__global__ void __launch_bounds__(256) k_spmm40(const float* __restrict__ p,
                                                const int* __restrict__ src,
                                                const int* __restrict__ dst,
                                                float* __restrict__ agg) {
  long gt = (long)blockIdx.x * 256 + threadIdx.x;
  int e = (int)(gt >> 5);
  int lane = threadIdx.x & 31;
  if (e >= NE) return;
  int s = src[e], d = dst[e];
  atomicAdd(&agg[(long)d * NC + lane], p[(long)s * NC + lane]);
  if (lane < NC - 32)
    atomicAdd(&agg[(long)d * NC + 32 + lane], p[(long)s * NC + 32 + lane]);
}

// ------- out = agg2*in_norm + b2 + h @ l2^T : bias+conv folded into C init -------
__global__ void __launch_bounds__(256) k_final(const float* __restrict__ h,
                                               const float* __restrict__ L2,
                                               const float* __restrict__ agg2,
                                               const float* __restrict__ innorm,
                                               const float* __restrict__ b2,
                                               float* __restrict__ out) {
  __shared__ float sW[48 * WPAD];
  stage_w40(L2, sW);
  __syncthreads();

  const int lane = threadIdx.x & 31;
  const int wave = threadIdx.x >> 5;
  const int rb = blockIdx.x * 8 + wave;
  if (rb * 16 >= NN) return;
  const int m0 = rb * 16;
  const int nlan = lane & 15;
  const int koff = (lane >> 4) * 2;
  const int hi8 = (lane >> 4) << 3;
  v8f acc[3];
#pragma unroll
  for (int t = 0; t < 3; ++t) {
    int col = t * 16 + nlan;
#pragma unroll
    for (int r = 0; r < 8; ++r) {
      int row = m0 + r + hi8;
      acc[t][r] = (col < NC) ? (agg2[(long)row * NC + col] * innorm[row] + b2[col]) : 0.f;
    }
  }
  const float* arow = h + (long)(m0 + nlan) * F + koff;
  for (int ks = 0; ks < 32; ++ks) {
    int k0 = ks * 4;
    float2 af = *(const float2*)(arow + k0);
    v2f a; a.x = af.x; a.y = af.y;
#pragma unroll
    for (int t = 0; t < 3; ++t) {
      float2 bf = *(const float2*)&sW[(t * 16 + nlan) * WPAD + koff + k0];
      v2f b; b.x = bf.x; b.y = bf.y;
      acc[t] = __builtin_amdgcn_wmma_f32_16x16x4_f32(false, a, false, b,
                                                     (short)0, acc[t], false, false);
    }
  }
#pragma unroll
  for (int t = 0; t < 3; ++t) {
    int col = t * 16 + nlan;
    if (col < NC) {
#pragma unroll
      for (int r = 0; r < 8; ++r)
        out[(long)(m0 + r + hi8) * NC + col] = acc[t][r];
    }
  }
}

extern "C" void kernel_launch(void* const* d_in, const int* in_sizes, int n_in,
                              void* d_out, int out_size, void* d_ws, size_t ws_size,
                              hipStream_t stream) {
  const float* feat  = (const float*)d_in[0];
  const int*   src   = (const int*)d_in[1];
  const int*   dst   = (const int*)d_in[2];
  const float* w0    = (const float*)d_in[3];
  const float* w1    = (const float*)d_in[4];
  const float* w2    = (const float*)d_in[5];
  const float* b2    = (const float*)d_in[6];
  const float* l0    = (const float*)d_in[7];
  const float* l1    = (const float*)d_in[8];
  const float* l2    = (const float*)d_in[9];
  const float* g0    = (const float*)d_in[10];
  const float* beta0 = (const float*)d_in[11];
  const float* g1    = (const float*)d_in[12];
  const float* beta1 = (const float*)d_in[13];
  float* out = (float*)d_out;

  float* ws = (float*)d_ws;
  float* norm_out = ws;                 // NN floats
  float* norm_in  = ws + NN;            // NN floats
  float* stats    = ws + 2 * NN;        // 256 floats
  const long BIG = (long)NN * F;        // 12.8M floats
  float* B0 = ws + 200704;
  float* B1 = B0 + BIG;
  float* B2 = B1 + BIG;

  const int eGrid  = (NE + 255) / 256;              // 6250
  const int sGrid  = (NN * (F / 4) + 255) / 256;    // 12500
  const int spGrid = (int)(((long)NE * 32) / 256);  // 200000
  const int gGrid  = (NN / 16 + 7) / 8;             // 782
  const int bGrid  = (int)(((long)NN * F + 255) / 256); // 50000

  // degrees -> norms
  hipMemsetAsync(norm_out, 0, 2 * NN * sizeof(float), stream);
  k_degree<<<eGrid, 256, 0, stream>>>(src, dst, norm_out, norm_in);
  k_norm<<<(2 * NN + 255) / 256, 256, 0, stream>>>(norm_out);

  // ---- layer 0: h = feat ----
  k_scale<<<sGrid, 256, 0, stream>>>(feat, norm_out, B0);        // hs -> B0
  hipMemsetAsync(B1, 0, BIG * sizeof(float), stream);
  k_spmm128<<<spGrid, 256, 0, stream>>>(B0, src, dst, B1);       // agg -> B1
  k_gemm_z<<<gGrid, 256, 0, stream>>>(B1, feat, w0, l0, norm_in, B2); // z -> B2
  hipMemsetAsync(stats, 0, 256 * sizeof(float), stream);
  k_colstats<<<400, 128, 0, stream>>>(B2, stats);
  k_bnrelu<<<bGrid, 256, 0, stream>>>(B2, stats, g0, beta0, B0); // h1 -> B0

  // ---- layer 1: h = B0 ----
  k_scale<<<sGrid, 256, 0, stream>>>(B0, norm_out, B1);          // hs -> B1
  hipMemsetAsync(B2, 0, BIG * sizeof(float), stream);
  k_spmm128<<<spGrid, 256, 0, stream>>>(B1, src, dst, B2);       // agg -> B2
  k_gemm_z<<<gGrid, 256, 0, stream>>>(B2, B0, w1, l1, norm_in, B1);   // z -> B1
  hipMemsetAsync(stats, 0, 256 * sizeof(float), stream);
  k_colstats<<<400, 128, 0, stream>>>(B1, stats);
  k_bnrelu<<<bGrid, 256, 0, stream>>>(B1, stats, g1, beta1, B2); // h2 -> B2

  // ---- layer 2: h = B2, project-first (128 -> 40) ----
  k_scale<<<sGrid, 256, 0, stream>>>(B2, norm_out, B0);          // hs -> B0
  k_gemm_p<<<gGrid, 256, 0, stream>>>(B0, w2, B1);               // p -> B1 (NN x 40)
  hipMemsetAsync(B0, 0, (long)NN * NC * sizeof(float), stream);
  k_spmm40<<<spGrid, 256, 0, stream>>>(B1, src, dst, B0);        // agg2 -> B0
  k_final<<<gGrid, 256, 0, stream>>>(B2, l2, B0, norm_in, b2, out);
}